// PDFSampler_90649579749983
// MI455X (gfx1250) — compile-verified
//
#include <hip/hip_runtime.h>
#include <cstdint>

#define N_RAYS   131072
#define N_COARSE 64
#define N_SAMP   128
#define N_ALL    (N_COARSE + N_SAMP)   // 192
#define RPB      8                     // rays per block (1 wave32 per ray)
#define BLOCK    256                   // 8 waves

typedef unsigned int u32x4 __attribute__((ext_vector_type(4)));
typedef int          i32x8 __attribute__((ext_vector_type(8)));
typedef int          i32x4 __attribute__((ext_vector_type(4)));

// LDS aperture rule: flat address of a __shared__ object has the LDS byte
// address in its low 32 bits (addr[31:0] truncated).
__device__ __forceinline__ uint32_t lds_addr_of(const void* p) {
    return (uint32_t)(uintptr_t)p;
}

// Issue a TDM DMA: ndwords consecutive f32 from global -> LDS (1-row tile).
__device__ __forceinline__ void tdm_load_1d(uint32_t lds_byte_addr,
                                            const void* gptr, int ndwords) {
    uint64_t ga = (uint64_t)(uintptr_t)gptr;
    u32x4 g0;
    g0.x = 1u;                                   // count=1, user descriptor
    g0.y = lds_byte_addr;                        // lds_addr [63:32]
    g0.z = (uint32_t)(ga & 0xFFFFFFFFu);         // global_addr lo
    g0.w = (uint32_t)((ga >> 32) & 0x01FFFFFFu)  // global_addr [56:32]
         | (2u << 30);                           // type=2 ("image")
    i32x8 g1;
    g1[0] = (2 << 16);                                        // data_size=4B
    g1[1] = (int)(((uint32_t)ndwords & 0xFFFFu) << 16);       // tensor_dim0 lo16
    g1[2] = (int)(((uint32_t)ndwords >> 16) | (1u << 16));    // dim0 hi16 | tensor_dim1=1
    g1[3] = (int)(((uint32_t)ndwords & 0xFFFFu) << 16);       // tile_dim0
    g1[4] = 1;                                                // tile_dim1=1
    g1[5] = ndwords;                                          // tensor_dim0_stride lo32
    g1[6] = 0;
    g1[7] = 0;
    i32x4 z4 = {0, 0, 0, 0};                     // groups 2/3 unused (<=2D)
    i32x8 z8 = {0, 0, 0, 0, 0, 0, 0, 0};
    __builtin_amdgcn_tensor_load_to_lds(g0, g1, z4, z4, z8, 0);
}

__global__ __launch_bounds__(BLOCK) void PDFSampler_kernel(
    const float* __restrict__ rays_o, const float* __restrict__ rays_d,
    const float* __restrict__ z_vals, const float* __restrict__ weights,
    float* __restrict__ pts, float* __restrict__ z_all_out)
{
    __shared__ float s_zv  [RPB * 64];
    __shared__ float s_wt  [RPB * 64];
    __shared__ float s_cdf [RPB * 64];   // 63 used
    __shared__ float s_zmid[RPB * 64];   // 63 used
    __shared__ float s_zs  [RPB * 128];
    __shared__ float s_zall[RPB * 192];

    const int tid  = threadIdx.x;
    const int wave = tid >> 5;
    const int lane = tid & 31;
    const int ray0 = blockIdx.x * RPB;
    const int ray  = ray0 + wave;

    // ---- Phase A: one wave DMAs the whole block's tiles into LDS (TDM) ----
    if (tid == 0) {
        tdm_load_1d(lds_addr_of(s_zv), z_vals  + (size_t)ray0 * 64, RPB * 64);
        tdm_load_1d(lds_addr_of(s_wt), weights + (size_t)ray0 * 64, RPB * 64);
    }
    // small per-ray vectors fetched while the DMA is in flight
    const float ro0 = rays_o[(size_t)ray * 3 + 0];
    const float ro1 = rays_o[(size_t)ray * 3 + 1];
    const float ro2 = rays_o[(size_t)ray * 3 + 2];
    const float rd0 = rays_d[(size_t)ray * 3 + 0];
    const float rd1 = rays_d[(size_t)ray * 3 + 1];
    const float rd2 = rays_d[(size_t)ray * 3 + 2];

    __builtin_amdgcn_s_wait_tensorcnt(0);   // TENSORcnt -> 0 (no-op for waves 1..7)
    __syncthreads();

    float* zv   = s_zv   + wave * 64;
    float* wt   = s_wt   + wave * 64;
    float* cdf  = s_cdf  + wave * 64;
    float* zmid = s_zmid + wave * 64;
    float* zs   = s_zs   + wave * 128;
    float* zall = s_zall + wave * 192;

    // ---- Phase B: CDF over weights[1..62] via two wave32 inclusive scans ----
    float a = wt[lane + 1] + 1e-5f;                              // elems 1..32
    float b = (lane < 30) ? (wt[lane + 33] + 1e-5f) : 0.0f;      // elems 33..62
#pragma unroll
    for (int off = 1; off < 32; off <<= 1) {
        float ta = __shfl_up(a, off, 32);
        float tb = __shfl_up(b, off, 32);
        if (lane >= off) { a += ta; b += tb; }
    }
    const float totalA = __shfl(a, 31, 32);
    const float wsum   = totalA + __shfl(b, 31, 32);   // scan total == sum
    const float inv    = 1.0f / wsum;
    if (lane == 0) cdf[0] = 0.0f;
    cdf[lane + 1] = a * inv;
    if (lane < 30) cdf[lane + 33] = (totalA + b) * inv;

    // z_mid[0..62]
    zmid[lane] = 0.5f * (zv[lane] + zv[lane + 1]);
    if (lane < 31) zmid[lane + 32] = 0.5f * (zv[lane + 32] + zv[lane + 33]);
    __syncthreads();

    // ---- Phase C: 128 inverse-CDF samples (searchsorted right, 4/lane) ----
#pragma unroll
    for (int t = 0; t < 4; ++t) {
        const int s = lane + 32 * t;
        const float u = (float)s * (1.0f / 127.0f);
        int lo = 0, hi = 63;                       // first idx with cdf[idx] > u
        while (lo < hi) {
            int mid = (lo + hi) >> 1;
            if (cdf[mid] <= u) lo = mid + 1; else hi = mid;
        }
        const int below = (lo - 1 > 0) ? lo - 1 : 0;
        const int above = (lo < 62) ? lo : 62;
        const float cb = cdf[below], ca = cdf[above];
        const float bb = zmid[below], ba = zmid[above];
        float denom = ca - cb;
        if (denom < 1e-5f) denom = 1.0f;
        zs[s] = fmaf((u - cb) / denom, ba - bb, bb);
    }
    __syncthreads();

    // ---- Phase D: merge two sorted runs (64 + 128) via merge-path ranks ----
#pragma unroll
    for (int t = 0; t < 2; ++t) {                  // z_vals: strict lower bound in zs
        const int i = lane + 32 * t;
        const float v = zv[i];
        int lo = 0, hi = 128;
        while (lo < hi) { int mid = (lo + hi) >> 1; if (zs[mid] <  v) lo = mid + 1; else hi = mid; }
        zall[i + lo] = v;
    }
#pragma unroll
    for (int t = 0; t < 4; ++t) {                  // z_samples: upper bound in zv
        const int j = lane + 32 * t;
        const float v = zs[j];
        int lo = 0, hi = 64;
        while (lo < hi) { int mid = (lo + hi) >> 1; if (zv[mid] <= v) lo = mid + 1; else hi = mid; }
        zall[j + lo] = v;
    }
    __syncthreads();

    // ---- Phase E: stream out z_all and pts with non-temporal (TH=NT) stores:
    //      403 MB of write-once data must not displace the 70 MB input set in L2.
    float* pts_ray = pts       + (size_t)ray * (N_ALL * 3);
    float* zo_ray  = z_all_out + (size_t)ray * N_ALL;
#pragma unroll
    for (int t = 0; t < 6; ++t) {
        const int k = lane + 32 * t;
        const float z = zall[k];
        __builtin_nontemporal_store(z, zo_ray + k);
        __builtin_nontemporal_store(fmaf(rd0, z, ro0), pts_ray + k * 3 + 0);
        __builtin_nontemporal_store(fmaf(rd1, z, ro1), pts_ray + k * 3 + 1);
        __builtin_nontemporal_store(fmaf(rd2, z, ro2), pts_ray + k * 3 + 2);
    }
}

extern "C" void kernel_launch(void* const* d_in, const int* in_sizes, int n_in,
                              void* d_out, int out_size, void* d_ws, size_t ws_size,
                              hipStream_t stream) {
    const float* rays_o  = (const float*)d_in[0];
    const float* rays_d  = (const float*)d_in[1];
    const float* z_vals  = (const float*)d_in[2];
    const float* weights = (const float*)d_in[3];
    float* out   = (float*)d_out;
    float* pts   = out;                                   // [N, 192, 3]
    float* z_all = out + (size_t)N_RAYS * N_ALL * 3;      // [N, 192]
    dim3 grid(N_RAYS / RPB), block(BLOCK);
    hipLaunchKernelGGL(PDFSampler_kernel, grid, block, 0, stream,
                       rays_o, rays_d, z_vals, weights, pts, z_all);
}